// Net_59115929862916
// MI455X (gfx1250) — compile-verified
//
#include <hip/hip_runtime.h>
#include <hip/hip_bf16.h>

typedef __attribute__((ext_vector_type(2))) float v2f;
typedef __attribute__((ext_vector_type(8))) float v8f;

#define D 64
#define WSTRIDE 68   // padded LDS row stride (floats): 68 % 64 == 4 -> conflict-free b64 frag loads

// ---------------------------------------------------------------------------
// Transform: hr[r][n][:] = x[n][:] @ W[r]   (W row-major [3][64][64], K=d)
// One wave computes a 16x64 tile for relation blockIdx.y using
// V_WMMA_F32_16X16X4_F32 (exact fp32 — compute is not the bottleneck here).
// Weights staged TRANSPOSED in LDS (Wt[col][k], stride 68) so each B-fragment
// is one aligned ds_load_b64, bank-conflict-free across the wave.
// ---------------------------------------------------------------------------
__global__ __launch_bounds__(256)
void rgcn_transform(const float* __restrict__ x, const float* __restrict__ W,
                    float* __restrict__ hr, int ntiles, int nnodes) {
    __shared__ float Wt[D * WSTRIDE];
    const int r = blockIdx.y;
    // cooperative transposed stage of W[r]: Wt[col*WSTRIDE + kk] = W[r][kk][col]
    for (int i = threadIdx.x; i < D * D; i += 256) {
        int kk  = i >> 6;
        int col = i & 63;
        Wt[col * WSTRIDE + kk] = W[(size_t)r * D * D + i];
    }
    __syncthreads();

    const int lane = threadIdx.x & 31;
    const int wave = threadIdx.x >> 5;
    const int half = lane >> 4;   // 0: K pair {4k,4k+1}, 1: {4k+2,4k+3}
    const int lr   = lane & 15;
    const int tile = blockIdx.x * 8 + wave;
    if (tile >= ntiles) return;   // wave-uniform -> EXEC stays all-ones for WMMA

    const float* xrow = x + ((size_t)tile * 16 + lr) * D;

    v8f acc[4] = {};              // 4 column tiles of 16 -> full 16x64 output
#pragma unroll
    for (int k = 0; k < 16; ++k) {            // K advances by 4 per WMMA
        // A 16x4 f32 frag: lanes 0-15 rows M=0..15 K={4k,4k+1}; lanes 16-31 K={4k+2,4k+3}
        v2f a = *(const v2f*)(xrow + k * 4 + 2 * half);
#pragma unroll
        for (int nt = 0; nt < 4; ++nt) {
            const int col = nt * 16 + lr;
            // B 4x16 f32 frag: contiguous K pair in transposed LDS -> ds_load_b64
            v2f b = *(const v2f*)(&Wt[col * WSTRIDE + k * 4 + 2 * half]);
            acc[nt] = __builtin_amdgcn_wmma_f32_16x16x4_f32(
                false, a, false, b, (short)0, acc[nt], false, false);
        }
    }

    // C/D layout: VGPR v -> row v (lanes 0-15) / row v+8 (lanes 16-31), col = lane%16
    float* out = hr + ((size_t)r * nnodes + (size_t)tile * 16) * D;
#pragma unroll
    for (int nt = 0; nt < 4; ++nt)
#pragma unroll
        for (int v = 0; v < 8; ++v)
            out[(size_t)(v + half * 8) * D + nt * 16 + lr] = acc[nt][v];
}

// ---------------------------------------------------------------------------
// Edge scatter: agg[dst[e]] += hr[rel[e]][src[e]] * norm[e]
// 16 lanes per edge, float4 per lane (contiguous 256B gather per edge),
// hardware fp32 atomics into L2 (whole working set is L2-resident on MI455X).
// ---------------------------------------------------------------------------
__global__ __launch_bounds__(256)
void edge_agg(const float* __restrict__ hr, const int* __restrict__ esrc,
              const int* __restrict__ edst, const int* __restrict__ erel,
              const float* __restrict__ enorm, float* __restrict__ agg,
              int nedges, int nnodes) {
    long tid = (long)blockIdx.x * blockDim.x + threadIdx.x;
    int e = (int)(tid >> 4);
    int seg = (int)(tid & 15);
    if (e >= nedges) return;
    int s = esrc[e], d = edst[e], r = erel[e];
    float nm = enorm[e];
    const float4* p =
        (const float4*)(hr + ((size_t)r * nnodes + (size_t)s) * D) + seg;
    float4 m = *p;
    float* q = agg + (size_t)d * D + seg * 4;
    unsafeAtomicAdd(q + 0, m.x * nm);
    unsafeAtomicAdd(q + 1, m.y * nm);
    unsafeAtomicAdd(q + 2, m.z * nm);
    unsafeAtomicAdd(q + 3, m.w * nm);
}

// ---------------------------------------------------------------------------
// ReLU (+ optional skip)
// ---------------------------------------------------------------------------
__global__ __launch_bounds__(256)
void relu_skip(const float* __restrict__ agg, const float* __restrict__ skip,
               float* __restrict__ out, long n) {
    long i = (long)blockIdx.x * blockDim.x + threadIdx.x;
    if (i >= n) return;
    float v = agg[i] + (skip ? skip[i] : 0.0f);
    out[i] = fmaxf(v, 0.0f);
}

// ---------------------------------------------------------------------------
// Heads: xa = x@Wa.T + ba  (N x 2), xb = x@Wb.T + bb  (N x 21); out = concat
// ---------------------------------------------------------------------------
__global__ __launch_bounds__(256)
void heads(const float* __restrict__ x, const float* __restrict__ Wa,
           const float* __restrict__ ba, const float* __restrict__ Wb,
           const float* __restrict__ bb, float* __restrict__ out, int nnodes) {
    __shared__ float Ws[23 * D];
    __shared__ float bs[23];
    for (int i = threadIdx.x; i < 23 * D; i += 256)
        Ws[i] = (i < 2 * D) ? Wa[i] : Wb[i - 2 * D];
    if (threadIdx.x < 23)
        bs[threadIdx.x] = (threadIdx.x < 2) ? ba[threadIdx.x] : bb[threadIdx.x - 2];
    __syncthreads();

    int n = blockIdx.x * blockDim.x + threadIdx.x;
    if (n >= nnodes) return;
    float xr[D];
    const float4* xp = (const float4*)(x + (size_t)n * D);
#pragma unroll
    for (int i = 0; i < 16; ++i) {
        float4 t = xp[i];
        xr[4 * i + 0] = t.x; xr[4 * i + 1] = t.y;
        xr[4 * i + 2] = t.z; xr[4 * i + 3] = t.w;
    }
#pragma unroll 1
    for (int o = 0; o < 23; ++o) {
        float s = bs[o];
#pragma unroll
        for (int d = 0; d < D; ++d) s = fmaf(xr[d], Ws[o * D + d], s);
        if (o < 2) out[(size_t)n * 2 + o] = s;
        else       out[(size_t)nnodes * 2 + (size_t)n * 21 + (o - 2)] = s;
    }
}

extern "C" void kernel_launch(void* const* d_in, const int* in_sizes, int n_in,
                              void* d_out, int out_size, void* d_ws, size_t ws_size,
                              hipStream_t stream) {
    const float* v     = (const float*)d_in[0];
    const int*   esrc  = (const int*)  d_in[1];
    const int*   edst  = (const int*)  d_in[2];
    const int*   erel  = (const int*)  d_in[3];
    const float* enorm = (const float*)d_in[4];
    const float* W1    = (const float*)d_in[5];
    const float* W2    = (const float*)d_in[6];
    const float* W3    = (const float*)d_in[7];
    const float* Wa    = (const float*)d_in[8];
    const float* ba    = (const float*)d_in[9];
    const float* Wb    = (const float*)d_in[10];
    const float* bb    = (const float*)d_in[11];

    const int nnodes = in_sizes[0] / D;   // 100000
    const int nedges = in_sizes[1];       // 1200000
    const int ntiles = (nnodes + 15) / 16;

    float* ws  = (float*)d_ws;
    size_t NN  = (size_t)nnodes * D;
    float* hr  = ws;            // 3*NN
    float* agg = ws + 3 * NN;   // NN
    float* x0  = ws + 4 * NN;   // NN
    float* x1  = ws + 5 * NN;   // NN

    auto layer = [&](const float* xin, const float* W, const float* skip,
                     float* xout) {
        dim3 g((ntiles + 7) / 8, 3);
        rgcn_transform<<<g, 256, 0, stream>>>(xin, W, hr, ntiles, nnodes);
        hipMemsetAsync(agg, 0, NN * sizeof(float), stream);
        long tot = (long)nedges * 16;
        edge_agg<<<(tot + 255) / 256, 256, 0, stream>>>(hr, esrc, edst, erel,
                                                        enorm, agg, nedges, nnodes);
        relu_skip<<<((long)NN + 255) / 256, 256, 0, stream>>>(agg, skip, xout,
                                                              (long)NN);
    };

    layer(v,  W1, nullptr, x0);
    layer(x0, W2, x0,      x1);
    layer(x1, W3, x1,      x0);

    heads<<<(nnodes + 255) / 256, 256, 0, stream>>>(x0, Wa, ba, Wb, bb,
                                                    (float*)d_out, nnodes);
}